// FinePreprocess_56281251447262
// MI455X (gfx1250) — compile-verified
//
#include <hip/hip_runtime.h>

typedef __attribute__((ext_vector_type(2))) float v2f;
typedef __attribute__((ext_vector_type(4))) float v4f;
typedef __attribute__((ext_vector_type(8))) float v8f;

#define DF 128          // fine dim / output channels
#define DC 256          // coarse dim / merge input dim
#define HF 240
#define WFUL 320
#define WCOL 80         // coarse grid width
#define CH_STRIDE (HF*WFUL)   // 76800 floats between channels
#define LTOK 4800
#define NP 25           // 5x5 window positions

// B fragment packing: Bpack[pair][j][lane] = v4f
//   lane: col = lane&15, khalf = (lane>>4)*2, n = j*16+col, kb = 8*pair+khalf
//   v4f = { W[n][kb], W[n][kb+1], W[n][kb+4], W[n][kb+5] }
// -> one coalesced b128 load feeds TWO v_wmma_f32_16x16x4_f32 steps.

// ---------------------------------------------------------------------------
// Pack merge_W[:, :128] into fragment-major layout (16 k-pairs).
// ---------------------------------------------------------------------------
__global__ void pack_mergeW_kernel(const float* __restrict__ mergeW,
                                   v4f* __restrict__ BpackM) {
    int pr   = blockIdx.x;            // 0..15
    int t    = threadIdx.x;           // 0..255
    int lane = t & 31;
    int j    = t >> 5;                // 0..7
    int col  = lane & 15;
    int khalf = (lane >> 4) << 1;
    int n  = j * 16 + col;
    int kb = 8 * pr + khalf;
    const float* wr = mergeW + (size_t)n * DC;
    v4f q = { wr[kb], wr[kb + 1], wr[kb + 4], wr[kb + 5] };
    BpackM[(pr * 8 + j) * 32 + lane] = q;
}

// ---------------------------------------------------------------------------
// Fused context matrix, written directly in packed fragment layout (32 pairs):
//   Wf[n][k] = sum_c mergeW[n][128+c] * downW[c][k]
// Extra block computes bf[o] = mergeB[o] + sum_c mergeW[o][128+c]*downB[c].
// ---------------------------------------------------------------------------
__global__ void fuse_pack_kernel(const float* __restrict__ downW,
                                 const float* __restrict__ downB,
                                 const float* __restrict__ mergeW,
                                 const float* __restrict__ mergeB,
                                 v4f* __restrict__ BpackC,
                                 float* __restrict__ bfv) {
    int pr = blockIdx.x;              // 0..31 pack, 32 = bias
    int t  = threadIdx.x;
    if (pr < 32) {
        int lane = t & 31;
        int j    = t >> 5;
        int col  = lane & 15;
        int khalf = (lane >> 4) << 1;
        int n  = j * 16 + col;
        int kb = 8 * pr + khalf;
        float s0 = 0.f, s1 = 0.f, s2 = 0.f, s3 = 0.f;
        const float* mr = mergeW + (size_t)n * DC + DF;
        for (int c = 0; c < DF; ++c) {
            float mv = mr[c];
            const float* dr = downW + (size_t)c * DC;
            s0 += mv * dr[kb];
            s1 += mv * dr[kb + 1];
            s2 += mv * dr[kb + 4];
            s3 += mv * dr[kb + 5];
        }
        v4f q = { s0, s1, s2, s3 };
        BpackC[(pr * 8 + j) * 32 + lane] = q;
    } else if (t < DF) {
        float s = mergeB[t];
        for (int c = 0; c < DF; ++c)
            s += mergeW[(size_t)t * DC + DF + c] * downB[c];
        bfv[t] = s;
    }
}

// ---------------------------------------------------------------------------
// Kernel B: ctx[g][o] = bf[o] + sum_k c_feat[g][k] * Wf[o][k]   (g < 2M, K=256)
// One wave per 16x128 strip; 32 k-pairs, each: 2 A b64 loads + 8 coalesced
// b128 B loads (clause) + 16 WMMAs.
// ---------------------------------------------------------------------------
__global__ void ctx_kernel(const float* __restrict__ fc0,
                           const float* __restrict__ fc1,
                           const int* __restrict__ b_ids,
                           const int* __restrict__ i_ids,
                           const int* __restrict__ j_ids,
                           const v4f* __restrict__ BpackC,
                           const float* __restrict__ bfv,
                           float* __restrict__ ctx, int M) {
    int wave = (blockIdx.x * blockDim.x + threadIdx.x) >> 5;
    int lane = threadIdx.x & 31;
    int rows = 2 * M;
    int nstrips = (rows + 15) >> 4;
    if (wave >= nstrips) return;

    int rowin = lane & 15;
    int khalf = (lane >> 4) << 1;      // 0 (lanes 0-15) or 2 (lanes 16-31)
    int col   = lane & 15;

    int r = wave * 16 + rowin;
    bool rvalid = (r < rows);
    int rc = rvalid ? r : 0;           // clamp -> always-legal gather address
    int side = (rc >= M) ? 1 : 0;
    int m = rc - side * M;
    int b = b_ids[m];
    int idx = side ? j_ids[m] : i_ids[m];
    const float* fc = side ? fc1 : fc0;
    const float* arow = fc + ((size_t)b * LTOK + (size_t)idx) * DC + khalf;
    const v4f* bp = BpackC + lane;

    v8f acc[8] = {};
    for (int pr = 0; pr < DC / 8; ++pr) {
        v2f a01 = *(const v2f*)(arow);        // k = 8pr+khalf, +1
        v2f a23 = *(const v2f*)(arow + 4);    // k = 8pr+4+khalf, +1
        arow += 8;
        v4f bq[8];
#pragma unroll
        for (int j = 0; j < 8; ++j)
            bq[j] = bp[j * 32];               // coalesced b128
        bp += 256;
        a01.x = rvalid ? a01.x : 0.f;
        a01.y = rvalid ? a01.y : 0.f;
        a23.x = rvalid ? a23.x : 0.f;
        a23.y = rvalid ? a23.y : 0.f;
        asm volatile("" ::: "memory");        // keep loads grouped ahead of WMMAs
#pragma unroll
        for (int j = 0; j < 8; ++j) {
            v2f blo = bq[j].xy;
            v2f bhi = bq[j].zw;
            acc[j] = __builtin_amdgcn_wmma_f32_16x16x4_f32(
                false, a01, false, blo, (short)0, acc[j], false, false);
            acc[j] = __builtin_amdgcn_wmma_f32_16x16x4_f32(
                false, a23, false, bhi, (short)0, acc[j], false, false);
        }
    }

#pragma unroll
    for (int j = 0; j < 8; ++j) {
        int n = j * 16 + col;
        float bias = bfv[n];
#pragma unroll
        for (int v = 0; v < 8; ++v) {
            int rr = wave * 16 + v + ((lane >> 4) << 3);  // C/D: M = v (+8 upper half)
            if (rr < rows)
                ctx[(size_t)rr * DF + n] = acc[j][v] + bias;
        }
    }
}

// ---------------------------------------------------------------------------
// Kernel C: merged[r][o] = ctx[r/25][o] + sum_k f_win[r][k] * mergeW[o][k]
// r in [0, 2M*25), K = 128. One wave per 16x128 strip; 16 k-pairs, each:
// 4 scattered A b32 loads + 8 coalesced b128 B loads + 16 WMMAs.
// ---------------------------------------------------------------------------
__global__ void merge_kernel(const float* __restrict__ ff0,
                             const float* __restrict__ ff1,
                             const int* __restrict__ b_ids,
                             const int* __restrict__ i_ids,
                             const int* __restrict__ j_ids,
                             const v4f* __restrict__ BpackM,
                             const float* __restrict__ ctx,
                             float* __restrict__ out, int M) {
    int wave = (blockIdx.x * blockDim.x + threadIdx.x) >> 5;
    int lane = threadIdx.x & 31;
    long rows = (long)2 * M * NP;
    int nstrips = (int)((rows + 15) >> 4);
    if (wave >= nstrips) return;

    int rowin = lane & 15;
    int khalf = (lane >> 4) << 1;
    int col   = lane & 15;

    long r = (long)wave * 16 + rowin;
    bool rvalid = (r < rows);
    long rc = rvalid ? r : 0;
    int g = (int)(rc / NP);
    int p = (int)(rc - (long)g * NP);
    int side = (g >= M) ? 1 : 0;
    int m = g - side * M;
    int b = b_ids[m];
    int idx = side ? j_ids[m] : i_ids[m];
    const float* ff = side ? ff1 : ff0;

    int hi = idx / WCOL;
    int wi = idx - hi * WCOL;
    int py = p / 5;
    int px = p - py * 5;
    int h = hi * 4 - 2 + py;          // stride=4, pad=2
    int w = wi * 4 - 2 + px;
    bool valid = rvalid && (h >= 0) && (h < HF) && (w >= 0) && (w < WFUL);
    long off = (long)b * DF * CH_STRIDE + (long)h * WFUL + w;
    long off_safe = valid ? off : 0;  // clamp -> always-legal address
    const float* ap = ff + off_safe + (long)khalf * CH_STRIDE;
    const v4f* bp = BpackM + lane;

    v8f acc[8] = {};
    for (int pr = 0; pr < DF / 8; ++pr) {
        float t0 = ap[0];                       // k = 8pr+khalf
        float t1 = ap[CH_STRIDE];               // +1
        float t2 = ap[(long)4 * CH_STRIDE];     // k = 8pr+4+khalf
        float t3 = ap[(long)5 * CH_STRIDE];     // +1
        ap += (long)8 * CH_STRIDE;
        v4f bq[8];
#pragma unroll
        for (int j = 0; j < 8; ++j)
            bq[j] = bp[j * 32];                 // coalesced b128
        bp += 128;
        v2f a01, a23;
        a01.x = valid ? t0 : 0.f;
        a01.y = valid ? t1 : 0.f;
        a23.x = valid ? t2 : 0.f;
        a23.y = valid ? t3 : 0.f;
        asm volatile("" ::: "memory");          // keep loads grouped ahead of WMMAs
#pragma unroll
        for (int j = 0; j < 8; ++j) {
            v2f blo = bq[j].xy;
            v2f bhi = bq[j].zw;
            acc[j] = __builtin_amdgcn_wmma_f32_16x16x4_f32(
                false, a01, false, blo, (short)0, acc[j], false, false);
            acc[j] = __builtin_amdgcn_wmma_f32_16x16x4_f32(
                false, a23, false, bhi, (short)0, acc[j], false, false);
        }
    }

#pragma unroll
    for (int j = 0; j < 8; ++j) {
        int n = j * 16 + col;
#pragma unroll
        for (int v = 0; v < 8; ++v) {
            long rr = (long)wave * 16 + v + ((lane >> 4) << 3);
            if (rr < rows) {
                int gg = (int)(rr / NP);
                out[rr * DF + n] = acc[j][v] + ctx[(size_t)gg * DF + n];
            }
        }
    }
}

// ---------------------------------------------------------------------------
extern "C" void kernel_launch(void* const* d_in, const int* in_sizes, int n_in,
                              void* d_out, int out_size, void* d_ws, size_t ws_size,
                              hipStream_t stream) {
    const float* ff0    = (const float*)d_in[0];   // feat_f0 (B,128,240,320)
    const float* ff1    = (const float*)d_in[1];   // feat_f1
    const float* fc0    = (const float*)d_in[2];   // feat_c0 (B,4800,256)
    const float* fc1    = (const float*)d_in[3];   // feat_c1
    // d_in[4] = hw0_f, d_in[5] = hw0_c (scalars; geometry baked into constants)
    const int*   b_ids  = (const int*)d_in[6];
    const int*   i_ids  = (const int*)d_in[7];
    const int*   j_ids  = (const int*)d_in[8];
    const float* downW  = (const float*)d_in[9];   // (128,256)
    const float* downB  = (const float*)d_in[10];  // (128,)
    const float* mergeW = (const float*)d_in[11];  // (128,256)
    const float* mergeB = (const float*)d_in[12];  // (128,)
    float* out = (float*)d_out;

    int M = in_sizes[6];                           // 3000

    // workspace layout (floats):
    //   BpackC : 32*8*32*4 = 32768   (packed fused context matrix)
    //   BpackM : 16*8*32*4 = 16384   (packed merge_W[:, :128])
    //   bfv    : 128
    //   ctx    : 2M*128               (~3.27 MB total)
    float* wsf    = (float*)d_ws;
    v4f*   BpackC = (v4f*)wsf;
    v4f*   BpackM = (v4f*)(wsf + 32768);
    float* bfv    = wsf + 32768 + 16384;
    float* ctx    = bfv + DF;

    pack_mergeW_kernel<<<16, 256, 0, stream>>>(mergeW, BpackM);
    fuse_pack_kernel<<<33, 256, 0, stream>>>(downW, downB, mergeW, mergeB,
                                             BpackC, bfv);

    int strips_b = (2 * M + 15) >> 4;
    int blocks_b = (strips_b + 7) >> 3;            // 8 waves / 256-thread block
    ctx_kernel<<<blocks_b, 256, 0, stream>>>(fc0, fc1, b_ids, i_ids, j_ids,
                                             BpackC, bfv, ctx, M);

    long rows = (long)2 * M * NP;
    int strips_c = (int)((rows + 15) >> 4);
    int blocks_c = (strips_c + 7) >> 3;
    merge_kernel<<<blocks_c, 256, 0, stream>>>(ff0, ff1, b_ids, i_ids, j_ids,
                                               BpackM, ctx, out, M);
}